// LCN_19688130084922
// MI455X (gfx1250) — compile-verified
//
#include <hip/hip_runtime.h>

// Problem constants: B=64, C=1, H=W=1024, KS=16, F=128, OUT=10, L=4096
#define B_   64
#define H_   1024
#define W_   1024
#define F_   128
#define L_   4096
#define OUT_ 10

typedef __attribute__((ext_vector_type(2))) float    v2f;
typedef __attribute__((ext_vector_type(8))) float    v8f;
typedef __attribute__((ext_vector_type(4))) unsigned v4u;
typedef __attribute__((ext_vector_type(8))) unsigned v8u;

// LDS row strides (floats), padded: stride%64==4 -> 16 row-strided lanes hit
// distinct banks; stride*4 % 16 == 0 keeps rows 16B aligned.
#define PSTR 260
#define WSTR 260
#define YSTR 132

// Flat LDS layout (byte offsets) -- single __shared__ array so TDM lds_addr
// (offset from workgroup LDS base, which is 0 for the sole static alloc)
// matches pointer arithmetic below.
#define OFF_P   0u                       // 64 patches * 260 f32  = 66560 B
#define OFF_W   66560u                   // 128 rows   * 260 f32  = 133120 B (2 chunks of 64 rows)
#define OFF_WCH 66560u                   // bytes per 64-row W chunk
#define OFF_Y   199680u                  // 64 * 132 f32          = 33792 B
#define OFF_DW  233472u                  // 128 * 16 f32          = 8192 B
#define OFF_B   241664u                  // 128 f32               = 512 B
#define SMEM_BYTES 242176u

// D# group1 word0 flags: data_size=4B (2<<16), pad_enable (1<<20),
// pad_interval=256 DWORDs (7<<22), pad_amount=4 DWORDs (3<<25)
// -> hardware reproduces the 260-float padded row stride in LDS.
#define TDM_FLAGS ((2u << 16) | (1u << 20) | (7u << 22) | (3u << 25))

// ---------------------------------------------------------------------------
// One workgroup (8 wave32) per patch location l.
//   TDM:  P (3D 16x16x64 tile) and W (two 2D 256x64 tiles) DMA'd into LDS.
//   Phase c=0,1: wait TENSORcnt prefix, 16 WMMA f32 tiles over f-chunk c
//                (W chunk 1 DMA overlaps phase-0 compute).
//   Phase 2: fused decoder GEMM partial[64,16] = Y[64,128] x DW[128,16].
// ---------------------------------------------------------------------------
__global__ __launch_bounds__(256) void lcn_main_kernel(
    const float* __restrict__ x,     // [64,1,1024,1024]
    const float* __restrict__ wgt,   // [128,4096,1,16,16]
    const float* __restrict__ bias,  // [128,4096,1]
    const float* __restrict__ decw,  // [10, 128*4096]
    float* __restrict__ ws)          // [4096,64,10] partials
{
    __shared__ __align__(16) float smem[SMEM_BYTES / 4];
    float* Pl  = smem + OFF_P  / 4;
    float* Wl  = smem + OFF_W  / 4;
    float* Yl  = smem + OFF_Y  / 4;
    float* DWl = smem + OFF_DW / 4;
    float* Bl  = smem + OFF_B  / 4;

    const int l   = blockIdx.x;
    const int tid = threadIdx.x;
    const int r0  = (l >> 6) << 4;
    const int c0  = (l & 63) << 4;

    const int wave = tid >> 5;
    const int lane = tid & 31;
    const int hl   = lane & 15;
    const int hi   = lane >> 4;
    const int mt   = wave & 3;             // m-tile (b block = mt*16)
    const int nn   = (wave >> 2) << 1;     // n-tile pair within phase: 0 or 2

    // ---- Issue TDM descriptors (wave 0; TDM ignores EXEC, wave-level op) ----
    if (wave == 0) {
        // P: 3D tile 16(x) x 16(y,row stride 1024) x 64(z,batch stride 2^20)
        unsigned long long pa =
            (unsigned long long)x + ((unsigned long long)(r0 * W_ + c0) << 2);
        v4u pg0 = { 1u, OFF_P, (unsigned)pa,
                    ((unsigned)(pa >> 32) & 0x01FFFFFFu) | (2u << 30) };
        v8u pg1 = { TDM_FLAGS,
                    16u << 16,            // tensor_dim0 = 16
                    16u << 16,            // tensor_dim1 = 16
                    16u << 16,            // tile_dim0   = 16
                    16u | (64u << 16),    // tile_dim1=16, tile_dim2=64
                    1024u,                // tensor_dim0_stride = 1024 elems
                    0u,                   // dim0_stride hi / dim1_stride lo
                    16u };                // tensor_dim1_stride = 2^20 -> hi32 = 16
        v4u pg2 = { 64u, 0u, 0u, 0u };    // tensor_dim2 = 64
        v4u pg3 = { 0u, 0u, 0u, 0u };
        asm volatile("tensor_load_to_lds %0, %1, %2, %3"
                     :: "s"(pg0), "s"(pg1), "s"(pg2), "s"(pg3) : "memory");

        // W chunks: 2D tiles 256(x) x 64(y, row stride L*256 = 2^20 elems)
        v8u wg1 = { TDM_FLAGS,
                    256u << 16,           // tensor_dim0 = 256
                    64u << 16,            // tensor_dim1 = 64
                    256u << 16,           // tile_dim0   = 256
                    64u,                  // tile_dim1 = 64, tile_dim2 = 0
                    1048576u,             // tensor_dim0_stride = 2^20 elems
                    0u, 0u };
#pragma unroll
        for (int c = 0; c < 2; ++c) {
            unsigned long long wa = (unsigned long long)wgt +
                (((unsigned long long)(c * 64) * (unsigned long long)(L_ * 256) +
                  (unsigned long long)l * 256ull) << 2);
            v4u wg0 = { 1u, OFF_W + (unsigned)c * OFF_WCH, (unsigned)wa,
                        ((unsigned)(wa >> 32) & 0x01FFFFFFu) | (2u << 30) };
            asm volatile("tensor_load_to_lds %0, %1"
                         :: "s"(wg0), "s"(wg1) : "memory");
        }
    }

    // ---- small gathers with regular loads (decoder slice + bias) ----
    for (int i = tid; i < 128 * 16; i += 256) {
        const int f = i >> 4;
        const int o = i & 15;
        DWl[i] = (o < OUT_)
                   ? decw[(size_t)o * (size_t)(F_ * L_) + (size_t)f * L_ + l]
                   : 0.0f;
    }
    if (tid < 128) Bl[tid] = bias[tid * L_ + l];

    // ---- Phase 0/1: WMMA GEMM, f-chunk c; chunk-1 DMA overlaps phase-0 ----
    for (int c = 0; c < 2; ++c) {
        if (wave == 0) {
            if (c == 0) __builtin_amdgcn_s_wait_tensorcnt(1); // P + W0 done
            else        __builtin_amdgcn_s_wait_tensorcnt(0); // W1 done
        }
        __syncthreads();

        const int nb = (c << 2) + nn;     // global n-tiles: nb, nb+1
        v8f acc[2] = {};
        for (int k0 = 0; k0 < 256; k0 += 4) {
            // A frag (16x4 f32): lanes 0-15 K=k0,k0+1; lanes 16-31 K=k0+2,k0+3
            v2f a = *(const v2f*)&Pl[(mt * 16 + hl) * PSTR + k0 + 2 * hi];
#pragma unroll
            for (int j = 0; j < 2; ++j) {
                v2f b = *(const v2f*)&Wl[((nb + j) * 16 + hl) * WSTR + k0 + 2 * hi];
                acc[j] = __builtin_amdgcn_wmma_f32_16x16x4_f32(
                    false, a, false, b, (short)0, acc[j], false, false);
            }
        }
        // bias + relu, scatter tiles to Yl (half-wave writes row-contiguous)
#pragma unroll
        for (int j = 0; j < 2; ++j) {
            const int f  = (nb + j) * 16 + hl;
            const float bv = Bl[f];
#pragma unroll
            for (int r = 0; r < 8; ++r) {
                const float v = acc[j][r] + bv;
                Yl[(mt * 16 + hi * 8 + r) * YSTR + f] = fmaxf(v, 0.0f);
            }
        }
    }
    __syncthreads();

    // ---- Phase 2: decoder GEMM partial[64,16] = Y[64,128] x DW[128,16] ----
    if (wave < 4) {
        v8f a2 = {};
        for (int k0 = 0; k0 < 128; k0 += 4) {
            v2f a = *(const v2f*)&Yl[(wave * 16 + hl) * YSTR + k0 + 2 * hi];
            v2f b;
            b.x = DWl[(k0 + 2 * hi) * 16 + hl];
            b.y = DWl[(k0 + 2 * hi + 1) * 16 + hl];
            a2 = __builtin_amdgcn_wmma_f32_16x16x4_f32(
                false, a, false, b, (short)0, a2, false, false);
        }
        if (hl < OUT_) {
            float* dst = ws + (size_t)l * (B_ * OUT_);
#pragma unroll
            for (int r = 0; r < 8; ++r) {
                const int brow = wave * 16 + hi * 8 + r;
                dst[brow * OUT_ + hl] = a2[r];
            }
        }
    }
}

// ---------------------------------------------------------------------------
// Deterministic reduction: out[b,o] = dec_b[o] + sum_l ws[l,b,o]
// ---------------------------------------------------------------------------
__global__ __launch_bounds__(256) void lcn_reduce_kernel(
    const float* __restrict__ ws, const float* __restrict__ decb,
    float* __restrict__ out)
{
    const int bo = blockIdx.x;        // 0..639 = b*10 + o
    const int o  = bo % OUT_;
    __shared__ float s[256];

    float acc = 0.0f;
    for (int li = threadIdx.x; li < L_; li += 256)
        acc += ws[(size_t)li * (B_ * OUT_) + bo];
    s[threadIdx.x] = acc;
    __syncthreads();
#pragma unroll
    for (int st = 128; st > 0; st >>= 1) {
        if (threadIdx.x < st) s[threadIdx.x] += s[threadIdx.x + st];
        __syncthreads();
    }
    if (threadIdx.x == 0) out[bo] = s[0] + decb[o];
}

// ---------------------------------------------------------------------------
extern "C" void kernel_launch(void* const* d_in, const int* in_sizes, int n_in,
                              void* d_out, int out_size, void* d_ws, size_t ws_size,
                              hipStream_t stream)
{
    const float* x    = (const float*)d_in[0];
    const float* wgt  = (const float*)d_in[1];
    const float* bias = (const float*)d_in[2];
    const float* decw = (const float*)d_in[3];
    const float* decb = (const float*)d_in[4];
    float* ws  = (float*)d_ws;              // needs 4096*64*10*4 = 10.5 MB
    float* out = (float*)d_out;             // 640 floats

    lcn_main_kernel<<<L_, 256, 0, stream>>>(x, wgt, bias, decw, ws);
    lcn_reduce_kernel<<<B_ * OUT_, 256, 0, stream>>>(ws, decb, out);
}